// SparseEfficientNodeLevelAttention_69896297775436
// MI455X (gfx1250) — compile-verified
//
#include <hip/hip_runtime.h>
#include <math.h>

#define N_NODES 50000
#define N_EDGES 400000
#define HID 256
#define HEADS 4
#define OUT_CH 64
#define LRELU_ALPHA 0.2f
#define LN_EPS 1e-5f

typedef __attribute__((ext_vector_type(2))) float v2f;
typedef __attribute__((ext_vector_type(8))) float v8f;
typedef __attribute__((ext_vector_type(4))) unsigned u32x4;
typedef __attribute__((ext_vector_type(8))) int i32x8;
typedef __attribute__((ext_vector_type(4))) int i32x4;

// ---------- float <-> order-preserving uint (for atomic float max) ----------
__device__ __forceinline__ unsigned f2ord(float f) {
  unsigned b = __float_as_uint(f);
  return (b & 0x80000000u) ? ~b : (b | 0x80000000u);
}
__device__ __forceinline__ float ord2f(unsigned k) {
  unsigned b = (k & 0x80000000u) ? (k & 0x7FFFFFFFu) : ~k;
  return __uint_as_float(b);
}

// ============================================================================
// GEMM: C[rows x 256] = A[rows x 256] @ B[256 x 256]  (fp32 WMMA 16x16x4)
// Block = 256 threads = 8 waves. Block owns one 16-row tile; wave w owns
// columns [32w, 32w+32) as two 16x16 accumulator tiles.
// A tile (16x256 fp32, 16KB) staged into LDS by the Tensor Data Mover:
// wave 0 issues one tensor_load_to_lds with a 2-D D# descriptor, waits on
// TENSORcnt, then the workgroup barrier publishes the tile.
// EPI==1: += bias, ReLU (FFN epilogue).
// ============================================================================
template <int EPI>
__global__ __launch_bounds__(256) void gemm_n256(
    const float* __restrict__ A, const float* __restrict__ B,
    const float* __restrict__ bias, float* __restrict__ C) {
  __shared__ float atile[16 * HID];

  const int row0 = blockIdx.x * 16;

  if ((threadIdx.x >> 5) == 0) {  // wave 0: TDM stage of the A tile
    const unsigned lds_base = (unsigned)(uintptr_t)atile;
    const unsigned long long ga =
        (unsigned long long)(A + (size_t)row0 * HID);
    // ---- D# group 0: control | lds_addr | global_addr | type=2 ----
    u32x4 g0;
    g0[0] = 1u;  // count=1 (valid user descriptor), no gather, no restore
    g0[1] = lds_base;                          // bits  63:32 lds_addr
    g0[2] = (unsigned)(ga & 0xFFFFFFFFu);      // bits  95:64 global_addr lo
    g0[3] = (unsigned)((ga >> 32) & 0x01FFFFFFu) | (2u << 30);  // hi | type=2
    // ---- D# group 1: data_size=4B, 2-D tensor 256x16, tile 256x16 ----
    i32x8 g1;
    g1[0] = 0x00020000;        // workgroup_mask=0, data_size=2 (4 bytes)
    g1[1] = (int)(256u << 16); // tensor_dim0[15:0]=256 (abar addr=0)
    g1[2] = (int)(16u << 16);  // tensor_dim0 hi=0, tensor_dim1[15:0]=16
    g1[3] = (int)(256u << 16); // tensor_dim1 hi=0, tile_dim0=256
    g1[4] = 16;                // tile_dim1=16, tile_dim2=0
    g1[5] = 256;               // tensor_dim0_stride lo32 = 256 elements
    g1[6] = 0;                 // stride hi, tensor_dim1_stride lo
    g1[7] = 0;                 // tensor_dim1_stride hi
    i32x4 z4 = {0, 0, 0, 0};   // groups 2/3 unused (2-D tensor)
    i32x8 z8 = {0, 0, 0, 0, 0, 0, 0, 0};
    __builtin_amdgcn_tensor_load_to_lds(g0, g1, z4, z4, z8, 0);
    __builtin_amdgcn_s_wait_tensorcnt(0);
  }
  __syncthreads();

  const int wave = threadIdx.x >> 5;
  const int lane = threadIdx.x & 31;
  const int hi   = lane >> 4;   // lane half: 0 -> K+0..1, 1 -> K+2..3
  const int m    = lane & 15;
  const int n0   = wave * 32 + m;
  const int n1   = n0 + 16;

  v8f acc0 = {};
  v8f acc1 = {};
  for (int kb = 0; kb < HID; kb += 4) {
    const int ka = kb + 2 * hi;
    // A fragment: A[m][ka], A[m][ka+1] (16x4 layout)
    v2f a = *(const v2f*)&atile[m * HID + ka];
    // B fragments: B[ka..ka+1][n] (4x16 layout, rows split across lane halves)
    v2f b0, b1;
    b0.x = B[(size_t)ka * HID + n0];
    b0.y = B[(size_t)(ka + 1) * HID + n0];
    b1.x = B[(size_t)ka * HID + n1];
    b1.y = B[(size_t)(ka + 1) * HID + n1];
    acc0 = __builtin_amdgcn_wmma_f32_16x16x4_f32(false, a, false, b0,
                                                 (short)0, acc0, false, false);
    acc1 = __builtin_amdgcn_wmma_f32_16x16x4_f32(false, a, false, b1,
                                                 (short)0, acc1, false, false);
  }

#pragma unroll
  for (int r = 0; r < 8; ++r) {
    const int row = row0 + r + 8 * hi;  // C/D layout: VGPR r -> M = r + 8*half
    float v0 = acc0[r], v1 = acc1[r];
    if (EPI) {
      v0 = fmaxf(v0 + bias[n0], 0.f);
      v1 = fmaxf(v1 + bias[n1], 0.f);
    }
    C[(size_t)row * HID + n0] = v0;
    C[(size_t)row * HID + n1] = v1;
  }
}

// ============================================================================
// Per-node attention scalars: s_src[n,h] = <Wh[n,h,:], a_src[h]>, same for dst.
// One wave32 per node; lane l covers cols [8l, 8l+8) (head = l>>3).
// ============================================================================
__global__ __launch_bounds__(256) void node_scores(
    const float* __restrict__ Wh, const float* __restrict__ attn,
    float* __restrict__ ssrc, float* __restrict__ sdst) {
  const int wave = threadIdx.x >> 5;
  const int lane = threadIdx.x & 31;
  const int node = blockIdx.x * 8 + wave;
  if (node >= N_NODES) return;

  const int head = lane >> 3;
  const int c0 = lane * 8;
  const int d0 = c0 - head * OUT_CH;
  const float* w  = Wh + (size_t)node * HID + c0;
  const float* as = attn + head * 2 * OUT_CH + d0;
  const float* ad = as + OUT_CH;

  float ps = 0.f, pd = 0.f;
#pragma unroll
  for (int j = 0; j < 8; ++j) {
    const float v = w[j];
    ps += v * as[j];
    pd += v * ad[j];
  }
#pragma unroll
  for (int off = 1; off < 8; off <<= 1) {
    ps += __shfl_xor(ps, off, 32);
    pd += __shfl_xor(pd, off, 32);
  }
  if ((lane & 7) == 0) {
    ssrc[node * HEADS + head] = ps;
    sdst[node * HEADS + head] = pd;
  }
}

// ============================================================================
// Edge pass 1: e[e,h] = leaky_relu(s_src[row,h] + s_dst[col,h]);
// track global per-head max (LDS ds_max_u32 reduce -> global atomicMax).
// ============================================================================
__device__ __forceinline__ float lrelu(float x) {
  return x > 0.f ? x : LRELU_ALPHA * x;
}

__global__ __launch_bounds__(256) void edge_scores(
    const int* __restrict__ erow, const int* __restrict__ ecol,
    const float* __restrict__ ssrc, const float* __restrict__ sdst,
    float* __restrict__ ebuf, unsigned* __restrict__ hmax) {
  __shared__ unsigned smax[HEADS];
  if (threadIdx.x < HEADS) smax[threadIdx.x] = 0u;  // ord key 0 == -inf-ish
  __syncthreads();

  const int e = blockIdx.x * 256 + threadIdx.x;
  if (e < N_EDGES) {
    const int r = erow[e], c = ecol[e];
    const float4 s = *(const float4*)&ssrc[r * HEADS];
    const float4 d = *(const float4*)&sdst[c * HEADS];
    float4 v;
    v.x = lrelu(s.x + d.x);
    v.y = lrelu(s.y + d.y);
    v.z = lrelu(s.z + d.z);
    v.w = lrelu(s.w + d.w);
    *(float4*)&ebuf[(size_t)e * HEADS] = v;
    atomicMax(&smax[0], f2ord(v.x));
    atomicMax(&smax[1], f2ord(v.y));
    atomicMax(&smax[2], f2ord(v.z));
    atomicMax(&smax[3], f2ord(v.w));
  }
  __syncthreads();
  if (threadIdx.x < HEADS) atomicMax(&hmax[threadIdx.x], smax[threadIdx.x]);
}

// ============================================================================
// Edge pass 2: e <- exp(e - head_max); segment-sum into e_sum[row,h].
// ============================================================================
__global__ __launch_bounds__(256) void edge_exp(
    const int* __restrict__ erow, float* __restrict__ ebuf,
    const unsigned* __restrict__ hmax, float* __restrict__ esum) {
  const int e = blockIdx.x * 256 + threadIdx.x;
  if (e >= N_EDGES) return;
  const int r = erow[e];
  const float m0 = ord2f(hmax[0]), m1 = ord2f(hmax[1]);
  const float m2 = ord2f(hmax[2]), m3 = ord2f(hmax[3]);
  float4 v = *(float4*)&ebuf[(size_t)e * HEADS];
  v.x = __expf(v.x - m0);
  v.y = __expf(v.y - m1);
  v.z = __expf(v.z - m2);
  v.w = __expf(v.w - m3);
  *(float4*)&ebuf[(size_t)e * HEADS] = v;
  atomicAdd(&esum[r * HEADS + 0], v.x);
  atomicAdd(&esum[r * HEADS + 1], v.y);
  atomicAdd(&esum[r * HEADS + 2], v.z);
  atomicAdd(&esum[r * HEADS + 3], v.w);
}

// ============================================================================
// Edge pass 3: accum[row] += alpha[e,h] * Wh[col].  One wave32 per edge;
// lane l handles cols [8l, 8l+8) -> 8 fp32 global atomics per lane.
// ============================================================================
__global__ __launch_bounds__(256) void edge_aggregate(
    const int* __restrict__ erow, const int* __restrict__ ecol,
    const float* __restrict__ ebuf, const float* __restrict__ esum,
    const float* __restrict__ Wh, float* __restrict__ accum) {
  const int wave = threadIdx.x >> 5;
  const int lane = threadIdx.x & 31;
  const int e = blockIdx.x * 8 + wave;
  if (e >= N_EDGES) return;

  const int r = erow[e], c = ecol[e];
  const int head = lane >> 3;
  const float alpha =
      ebuf[(size_t)e * HEADS + head] / (esum[r * HEADS + head] + 1e-15f);

  const int c0 = lane * 8;
  const float4 w0 = *(const float4*)&Wh[(size_t)c * HID + c0];
  const float4 w1 = *(const float4*)&Wh[(size_t)c * HID + c0 + 4];
  float* dst = accum + (size_t)r * HID + c0;
  atomicAdd(dst + 0, w0.x * alpha);
  atomicAdd(dst + 1, w0.y * alpha);
  atomicAdd(dst + 2, w0.z * alpha);
  atomicAdd(dst + 3, w0.w * alpha);
  atomicAdd(dst + 4, w1.x * alpha);
  atomicAdd(dst + 5, w1.y * alpha);
  atomicAdd(dst + 6, w1.z * alpha);
  atomicAdd(dst + 7, w1.w * alpha);
}

// ============================================================================
// Residual + LayerNorm, in place into accum. One wave32 per node.
// ============================================================================
__global__ __launch_bounds__(256) void layer_norm(
    float* __restrict__ accum, const float* __restrict__ Wh,
    const float* __restrict__ gamma, const float* __restrict__ beta) {
  const int wave = threadIdx.x >> 5;
  const int lane = threadIdx.x & 31;
  const int node = blockIdx.x * 8 + wave;
  if (node >= N_NODES) return;

  const int c0 = lane * 8;
  const float* acc = accum + (size_t)node * HID;
  const float* wh = Wh + (size_t)node * HID;
  float v[8];
  float s = 0.f, ss = 0.f;
#pragma unroll
  for (int j = 0; j < 8; ++j) {
    v[j] = acc[c0 + j] + wh[c0 + j];
    s += v[j];
    ss += v[j] * v[j];
  }
#pragma unroll
  for (int off = 1; off < 32; off <<= 1) {
    s += __shfl_xor(s, off, 32);
    ss += __shfl_xor(ss, off, 32);
  }
  const float mean = s * (1.f / HID);
  const float var = ss * (1.f / HID) - mean * mean;
  const float rstd = rsqrtf(var + LN_EPS);
  float* out = accum + (size_t)node * HID;
#pragma unroll
  for (int j = 0; j < 8; ++j)
    out[c0 + j] = (v[j] - mean) * rstd * gamma[c0 + j] + beta[c0 + j];
}

// ============================================================================
extern "C" void kernel_launch(void* const* d_in, const int* in_sizes, int n_in,
                              void* d_out, int out_size, void* d_ws,
                              size_t ws_size, hipStream_t stream) {
  const float* h     = (const float*)d_in[0];
  const int*   erow  = (const int*)d_in[1];
  const int*   ecol  = (const int*)d_in[2];
  const float* W     = (const float*)d_in[3];
  const float* attn  = (const float*)d_in[4];
  const float* gamma = (const float*)d_in[5];
  const float* beta  = (const float*)d_in[6];
  const float* ffnw  = (const float*)d_in[7];
  const float* ffnb  = (const float*)d_in[8];
  float* out = (float*)d_out;

  // workspace layout (floats)
  float* ws    = (float*)d_ws;
  float* Wh    = ws;                           // N*256
  float* accum = Wh + (size_t)N_NODES * HID;   // N*256
  float* ebuf  = accum + (size_t)N_NODES * HID;  // E*4
  float* ssrc  = ebuf + (size_t)N_EDGES * HEADS; // N*4
  float* sdst  = ssrc + (size_t)N_NODES * HEADS; // N*4
  float* esum  = sdst + (size_t)N_NODES * HEADS; // N*4
  unsigned* hmax = (unsigned*)(esum + (size_t)N_NODES * HEADS);  // 4

  (void)hipMemsetAsync(accum, 0, (size_t)N_NODES * HID * sizeof(float), stream);
  (void)hipMemsetAsync(esum, 0, (size_t)N_NODES * HEADS * sizeof(float),
                       stream);
  (void)hipMemsetAsync(hmax, 0, HEADS * sizeof(unsigned), stream);

  gemm_n256<0><<<N_NODES / 16, 256, 0, stream>>>(h, W, nullptr, Wh);
  node_scores<<<N_NODES / 8, 256, 0, stream>>>(Wh, attn, ssrc, sdst);
  edge_scores<<<(N_EDGES + 255) / 256, 256, 0, stream>>>(erow, ecol, ssrc,
                                                         sdst, ebuf, hmax);
  edge_exp<<<(N_EDGES + 255) / 256, 256, 0, stream>>>(erow, ebuf, hmax, esum);
  edge_aggregate<<<N_EDGES / 8, 256, 0, stream>>>(erow, ecol, ebuf, esum, Wh,
                                                  accum);
  layer_norm<<<N_NODES / 8, 256, 0, stream>>>(accum, Wh, gamma, beta);
  gemm_n256<1><<<N_NODES / 16, 256, 0, stream>>>(accum, ffnw, ffnb, out);
}